// GraphConvolution_14568529068197
// MI455X (gfx1250) — compile-verified
//
#include <hip/hip_runtime.h>

typedef float v2f __attribute__((ext_vector_type(2)));
typedef float v8f __attribute__((ext_vector_type(8)));

#define NUM_NODES 50000
#define NUM_EDGES 640000
#define FEAT 128

// ---------------------------------------------------------------------------
// Kernel 1: zero the workspace (seg_sum [N*128] + counts [N]) as float4s.
// ---------------------------------------------------------------------------
__global__ void GraphConv_zero_ws(float4* __restrict__ ws, int n4) {
    int t = blockIdx.x * blockDim.x + threadIdx.x;
    if (t < n4) {
        ws[t] = make_float4(0.f, 0.f, 0.f, 0.f);
    }
}

// ---------------------------------------------------------------------------
// Kernel 2: edge aggregation. One wave (32 lanes) per edge; each lane handles
// 4 consecutive features (float4 gather + 4 f32 atomic scatters). Lane 0 adds
// the edge count. Edge indices are wave-uniform -> scalarized loads.
// ---------------------------------------------------------------------------
__global__ void GraphConv_aggregate(const float* __restrict__ feat,
                                    const int* __restrict__ edst,
                                    const int* __restrict__ esrc,
                                    float* __restrict__ seg,
                                    float* __restrict__ cnt) {
    int t = blockIdx.x * blockDim.x + threadIdx.x;
    int e = t >> 5;
    if (e >= NUM_EDGES) return;
    int lane = t & 31;

    int s = esrc[e];
    int d = edst[e];

    const float4 v = *(const float4*)&feat[(size_t)s * FEAT + lane * 4];
    float* dp = &seg[(size_t)d * FEAT + lane * 4];
    atomicAdd(dp + 0, v.x);
    atomicAdd(dp + 1, v.y);
    atomicAdd(dp + 2, v.z);
    atomicAdd(dp + 3, v.w);
    if (lane == 0) atomicAdd(&cnt[d], 1.0f);
}

// ---------------------------------------------------------------------------
// Kernel 3: fused dual GEMM + mean-normalize + concat + ReLU using
// V_WMMA_F32_16X16X4_F32. One wave computes one 16x16 tile of the [N,256]
// output. Tiles: 3125 row-tiles x 16 col-tiles (cols 0..127 = feat@W,
// 128..255 = aggregated@W). K=128 swept in steps of 4 -> 32 WMMAs/tile.
//
// Fragment layouts (ISA 7.12.2, fp32):
//   A 16x4 : lane l holds row m=l&15; VGPR0/1 = K = 2*(l>>4) + {0,1}
//   B 4x16 : lane l holds col n=l&15; VGPR0/1 = K = 2*(l>>4) + {0,1}
//   C/D    : VGPR r holds row m = r + 8*(l>>4), col n = l&15
// ---------------------------------------------------------------------------
__global__ void GraphConv_gemm_wmma(const float* __restrict__ feat,
                                    const float* __restrict__ seg,
                                    const float* __restrict__ cnt,
                                    const float* __restrict__ W,
                                    float* __restrict__ out) {
    const int lane  = threadIdx.x & 31;
    const int wave  = threadIdx.x >> 5;
    const int tile  = blockIdx.x * 8 + wave;        // 0 .. 49999
    const int col16 = tile & 15;                    // 16 col-tiles across [0,256)
    const int rowt  = tile >> 4;                    // 0 .. 3124
    const int half  = col16 >> 3;                   // 0: feat@W  1: aggregated@W
    const int wcol  = (col16 & 7) * 16;             // column base within W (0..112)
    const int rowb  = rowt * 16;

    const int m  = lane & 15;                       // A row / B col / D col
    const int kh = (lane >> 4) * 2;                 // K sub-offset for A/B frags

    const float* A = half ? seg : feat;
    float scale = 1.0f;
    if (half) {
        float c = cnt[rowb + m];
        scale = (c > 0.0f) ? (1.0f / c) : 0.0f;     // segment mean; empty -> 0
    }

    const float* arow = A + (size_t)(rowb + m) * FEAT + kh;
    const float* bcol = W + (size_t)kh * FEAT + wcol + m;

    v8f acc = {};
    #pragma unroll
    for (int k = 0; k < FEAT; k += 4) {
        const float2 av = *(const float2*)&arow[k];
        v2f a, b;
        a.x = av.x * scale;
        a.y = av.y * scale;
        b.x = bcol[(size_t)k * FEAT];
        b.y = bcol[(size_t)(k + 1) * FEAT];
        acc = __builtin_amdgcn_wmma_f32_16x16x4_f32(
            /*neg_a=*/false, a, /*neg_b=*/false, b,
            /*c_mod=*/(short)0, acc, /*reuse_a=*/false, /*reuse_b=*/false);
    }

    // Store with fused ReLU. D row = r + 8*(lane>>4), col = lane&15.
    const int out_col = half * FEAT + wcol + m;
    const int rb = rowb + (lane >> 4) * 8;
    #pragma unroll
    for (int r = 0; r < 8; ++r) {
        float v = acc[r];
        out[(size_t)(rb + r) * 256 + out_col] = v > 0.0f ? v : 0.0f;
    }
}

// ---------------------------------------------------------------------------
extern "C" void kernel_launch(void* const* d_in, const int* in_sizes, int n_in,
                              void* d_out, int out_size, void* d_ws, size_t ws_size,
                              hipStream_t stream) {
    const float* feat = (const float*)d_in[0];   // [N,128]
    const float* W    = (const float*)d_in[1];   // [128,128]
    const int*   edst = (const int*)d_in[2];     // [E]
    const int*   esrc = (const int*)d_in[3];     // [E]
    float* out = (float*)d_out;                  // [N,256]

    float* seg = (float*)d_ws;                   // [N*128]
    float* cnt = seg + (size_t)NUM_NODES * FEAT; // [N]

    // 1) zero seg_sum + counts (6,450,000 floats = 1,612,500 float4s)
    {
        int n4 = (NUM_NODES * FEAT + NUM_NODES) / 4;
        int blocks = (n4 + 255) / 256;
        GraphConv_zero_ws<<<blocks, 256, 0, stream>>>((float4*)d_ws, n4);
    }

    // 2) edge scatter-add (one wave per edge)
    {
        long long threads = (long long)NUM_EDGES * 32;
        int blocks = (int)((threads + 255) / 256);   // 80000
        GraphConv_aggregate<<<blocks, 256, 0, stream>>>(feat, edst, esrc, seg, cnt);
    }

    // 3) fused WMMA GEMMs + concat + ReLU (50000 tiles, 8 waves/block)
    {
        int tiles = (NUM_NODES / 16) * 16;           // 50000
        int blocks = tiles / 8;                      // 6250
        GraphConv_gemm_wmma<<<blocks, 256, 0, stream>>>(feat, seg, cnt, W, out);
    }
}